// FMNetOriginal_80135499809319
// MI455X (gfx1250) — compile-verified
//
#include <hip/hip_runtime.h>

typedef float v2f __attribute__((ext_vector_type(2)));
typedef float v8f __attribute__((ext_vector_type(8)));

#define LMBDA  100.0f
#define B_     8
#define N_     20000
#define K_     128
#define C_     128
#define NCHUNK 20
#define CHUNK  (N_ / NCHUNK)   // 1000, divisible by 4

// ---------------------------------------------------------------- zero init
__global__ void zero_kernel(float* __restrict__ p, int n) {
    int i = blockIdx.x * blockDim.x + threadIdx.x;
    if (i < n) p[i] = 0.0f;
}

// ---------------------------------------------------------------- stage 1
// A[b] = Ex[b] (KxN) @ Fx[b] (NxC), Bm likewise.
// One wave per 16x64 tile-strip per N-chunk: 4 independent v8f accumulators
// (4 WMMA dep chains, shared A-fragment), fp32 WMMA 16x16x4, partial tiles
// merged with global_atomic_add_f32.
__global__ void __launch_bounds__(256) proj_gemm_kernel(
    const float* __restrict__ feat_x, const float* __restrict__ feat_y,
    const float* __restrict__ etx,    const float* __restrict__ ety,
    float* __restrict__ Aout,         float* __restrict__ Bout) {
    int wid  = blockIdx.x * (blockDim.x >> 5) + (threadIdx.x >> 5);
    int lane = threadIdx.x & 31;

    int nc  = wid % NCHUNK;  int t = wid / NCHUNK;
    int cs  = t & 1;         t >>= 1;      // 64-wide column strip (0,1)
    int kt  = t & 7;         t >>= 3;      // 16-row k tile
    int b   = t & 7;         int mat = t >> 3;
    int k0  = kt << 4;
    int c0  = cs << 6;

    const float* E = (mat ? ety    : etx)    + (size_t)b * K_ * N_;
    const float* F = (mat ? feat_y : feat_x) + (size_t)b * N_ * C_;
    float*     Out = (mat ? Bout   : Aout)   + (size_t)b * K_ * C_;

    int m  = lane & 15;          // A row / B col within subtile
    int kh = lane >> 4;          // K-half selector

    const float* Erow = E + (size_t)(k0 + m) * N_ + 2 * kh;          // A-frag
    const float* Fc   = F + (size_t)2 * kh * C_ + c0 + m;            // B-frag base

    v8f acc0 = {}, acc1 = {}, acc2 = {}, acc3 = {};
    int n0 = nc * CHUNK, n1 = n0 + CHUNK;
    for (int n = n0; n < n1; n += 4) {
        v2f av = *(const v2f*)(Erow + n);            // K = 2kh, 2kh+1 (b64 load)
        const float* fp = Fc + (size_t)n * C_;
        v2f b0, b1, b2, b3;
        b0.x = fp[ 0];        b0.y = fp[ 0 + C_];    // subtile j: cols c0+16j
        b1.x = fp[16];        b1.y = fp[16 + C_];
        b2.x = fp[32];        b2.y = fp[32 + C_];
        b3.x = fp[48];        b3.y = fp[48 + C_];
        acc0 = __builtin_amdgcn_wmma_f32_16x16x4_f32(false, av, false, b0, (short)0, acc0, false, false);
        acc1 = __builtin_amdgcn_wmma_f32_16x16x4_f32(false, av, false, b1, (short)0, acc1, false, false);
        acc2 = __builtin_amdgcn_wmma_f32_16x16x4_f32(false, av, false, b2, (short)0, acc2, false, false);
        acc3 = __builtin_amdgcn_wmma_f32_16x16x4_f32(false, av, false, b3, (short)0, acc3, false, false);
    }

    float* o = Out + (size_t)(k0 + 8 * kh) * C_ + c0 + m;
    #pragma unroll
    for (int r = 0; r < 8; ++r) {
        unsafeAtomicAdd(o + (size_t)r * C_,      acc0[r]);  // global_atomic_add_f32
        unsafeAtomicAdd(o + (size_t)r * C_ + 16, acc1[r]);
        unsafeAtomicAdd(o + (size_t)r * C_ + 32, acc2[r]);
        unsafeAtomicAdd(o + (size_t)r * C_ + 48, acc3[r]);
    }
}

// ---------------------------------------------------------------- stage 2
// AAt[b] = A[b]·A[b]^T, BAt[b] = Bm[b]·A[b]^T. One wave per 16x16 tile.
__global__ void __launch_bounds__(256) gram_kernel(
    const float* __restrict__ A,  const float* __restrict__ Bm,
    float* __restrict__ AAt,      float* __restrict__ BAt) {
    int wid  = blockIdx.x * (blockDim.x >> 5) + (threadIdx.x >> 5);
    int lane = threadIdx.x & 31;
    int tile = wid & 63;  int t = wid >> 6;
    int b    = t & 7;     int mat = t >> 3;
    int k0   = (tile >> 3) << 4;
    int l0   = (tile & 7)  << 4;

    const float* P = (mat ? Bm : A) + (size_t)b * K_ * C_;
    const float* Q = A              + (size_t)b * K_ * C_;
    float*     Out = (mat ? BAt : AAt) + (size_t)b * K_ * K_;

    int m  = lane & 15;
    int kh = lane >> 4;
    const float* Prow = P + (size_t)(k0 + m) * C_ + 2 * kh;
    const float* Qrow = Q + (size_t)(l0 + m) * C_ + 2 * kh;  // transposed walk

    v8f acc = {};
    for (int c = 0; c < C_; c += 4) {
        v2f av = *(const v2f*)(Prow + c);
        v2f bv = *(const v2f*)(Qrow + c);
        acc = __builtin_amdgcn_wmma_f32_16x16x4_f32(
            false, av, false, bv, (short)0, acc, false, false);
    }
    float* o = Out + (size_t)(k0 + 8 * kh) * K_ + l0 + m;
    #pragma unroll
    for (int r = 0; r < 8; ++r)
        o[(size_t)r * K_] = acc[r];
}

// ---------------------------------------------------------------- mask D
__global__ void __launch_bounds__(128) mask_kernel(
    const float* __restrict__ evx, const float* __restrict__ evy,
    float* __restrict__ D) {
    __shared__ float red[128], sgx[128], sgy[128];
    int b = blockIdx.x, t = threadIdx.x;
    float ex = evx[b * K_ + t], ey = evy[b * K_ + t];
    red[t] = fmaxf(ex, ey);
    __syncthreads();
    for (int s = 64; s > 0; s >>= 1) {
        if (t < s) red[t] = fmaxf(red[t], red[t + s]);
        __syncthreads();
    }
    float inv = 1.0f / red[0];           // scale = max(max ex, max ey)
    sgx[t] = sqrtf(ex * inv);            // gamma = 0.5
    sgy[t] = sqrtf(ey * inv);
    __syncthreads();
    float g2 = sgy[t];
    float d2 = 1.0f / (g2 * g2 + 1.0f);
    float a2 = g2 * d2;
    float* Drow = D + ((size_t)b * K_ + t) * K_;
    for (int j = 0; j < K_; ++j) {
        float g1 = sgx[j];
        float d1 = 1.0f / (g1 * g1 + 1.0f);
        float re = a2 - g1 * d1;
        float im = d2 - d1;
        Drow[j]  = re * re + im * im;
    }
}

// ---------------------------------------------------------------- solve
// One workgroup per (b,i): Gauss-Jordan on [M | rhs] in 68KB dynamic LDS
// (legal on CDNA5's 320KB WGP LDS). M = AAt[b] + 100*diag(D[b,i]) is SPD
// -> no pivoting needed.
#define SSTRIDE 133   // odd: pivot-column reads are bank-conflict-free
__global__ void __launch_bounds__(128) solve_kernel(
    const float* __restrict__ AAt, const float* __restrict__ BAt,
    const float* __restrict__ D,   float* __restrict__ Cxy) {
    extern __shared__ float S[];
    int b = blockIdx.x >> 7, i = blockIdx.x & 127;
    int r = threadIdx.x;

    const float* Mrow = AAt + ((size_t)b * K_ + r) * K_;
    float* Srow = S + r * SSTRIDE;
    for (int c = 0; c < K_; ++c) Srow[c] = Mrow[c];
    Srow[r]  += LMBDA * D[((size_t)b * K_ + i) * K_ + r];
    Srow[K_]  = BAt[((size_t)b * K_ + i) * K_ + r];
    __syncthreads();

    for (int p = 0; p < K_; ++p) {
        float invp = 1.0f / S[p * SSTRIDE + p];
        float f    = Srow[p] * invp;
        if (r != p) {
            const float* Prow = S + p * SSTRIDE;
            for (int c = p; c <= K_; ++c)
                Srow[c] -= f * Prow[c];
        }
        __syncthreads();
    }
    Cxy[((size_t)b * K_ + i) * K_ + r] = Srow[K_] / Srow[r];
}

// ---------------------------------------------------------------- launch
extern "C" void kernel_launch(void* const* d_in, const int* in_sizes, int n_in,
                              void* d_out, int out_size, void* d_ws, size_t ws_size,
                              hipStream_t stream) {
    const float* feat_x  = (const float*)d_in[0];
    const float* feat_y  = (const float*)d_in[1];
    const float* evals_x = (const float*)d_in[2];
    const float* evals_y = (const float*)d_in[3];
    const float* etx     = (const float*)d_in[4];
    const float* ety     = (const float*)d_in[5];
    float* out = (float*)d_out;

    float* ws  = (float*)d_ws;
    float* A   = ws;                       // B*K*C
    float* Bm  = A   + B_ * K_ * C_;       // B*K*C
    float* AAt = Bm  + B_ * K_ * C_;       // B*K*K
    float* BAt = AAt + B_ * K_ * K_;       // B*K*K
    float* D   = BAt + B_ * K_ * K_;       // B*K*K

    // zero the atomic accumulators (A, Bm)
    int zn = 2 * B_ * K_ * C_;
    zero_kernel<<<(zn + 255) / 256, 256, 0, stream>>>(A, zn);

    // stage 1: 2 mats * 8 b * 8 ktiles * 2 cstrips * NCHUNK waves, 8 waves/block
    int waves1 = 2 * B_ * 8 * 2 * NCHUNK;  // 5120
    proj_gemm_kernel<<<waves1 / 8, 256, 0, stream>>>(feat_x, feat_y, etx, ety, A, Bm);

    // stage 2: 2 * 8 * 64 waves
    gram_kernel<<<(2 * B_ * 64) / 8, 256, 0, stream>>>(A, Bm, AAt, BAt);

    mask_kernel<<<B_, 128, 0, stream>>>(evals_x, evals_y, D);

    size_t lds = (size_t)K_ * SSTRIDE * sizeof(float);   // 68,096 B
    solve_kernel<<<B_ * K_, 128, lds, stream>>>(AAt, BAt, D, out);
}